// StabilizerEmbedder_15710990369420
// MI455X (gfx1250) — compile-verified
//
#include <hip/hip_runtime.h>

typedef float v2f __attribute__((ext_vector_type(2)));
typedef float v8f __attribute__((ext_vector_type(8)));

#define F 6
#define H 256
#define S_LEN 512
#define ROWS_PER_WAVE 16
#define WAVES_PER_BLOCK 8
#define ROWS_PER_BLOCK (ROWS_PER_WAVE * WAVES_PER_BLOCK)
#define LN_EPS 1e-5f

// One wave32 computes a 16-row x 256-col output slab entirely in registers:
//   h = x@W (two V_WMMA_F32_16X16X4_F32 per 16x16 tile, K=6 padded to 8)
//     + bsum + emb, then LayerNorm via per-VGPR sums + 16-lane shfl butterfly,
//   streamed out with non-temporal stores (output is 1 GB, don't keep in L2).
__global__ __launch_bounds__(256) void stab_embed_ln_kernel(
    const float* __restrict__ x, const float* __restrict__ W,
    const float* __restrict__ b, const float* __restrict__ emb,
    const float* __restrict__ gamma, const float* __restrict__ beta,
    float* __restrict__ out)
{
    __shared__ float sW[F * H];      // 6 KB: W staged once per block
    __shared__ float sBsum[H];       // sum_f b[f][h]
    __shared__ float sGamma[H];
    __shared__ float sBeta[H];

    const int tid = threadIdx.x;
    for (int i = tid; i < F * H; i += 256) sW[i] = W[i];
    {
        float s = 0.f;
#pragma unroll
        for (int f = 0; f < F; ++f) s += b[f * H + tid];
        sBsum[tid]  = s;
        sGamma[tid] = gamma[tid];
        sBeta[tid]  = beta[tid];
    }
    __syncthreads();

    const int wave = tid >> 5;
    const int lane = tid & 31;
    const int lo   = lane & 15;      // N position / M position for A
    const int hi   = lane >> 4;      // 0: K=0..1 (A/B), rows M=v; 1: K=2..3, rows M=v+8

    const size_t rowBase = (size_t)blockIdx.x * ROWS_PER_BLOCK
                         + (size_t)wave * ROWS_PER_WAVE;

    // ---- A tiles (16x4 f32 layout: lane<16 -> {K0,K1}, lane>=16 -> {K2,K3}) ----
    v2f a0, a1;
    {
        const float* xp = x + (rowBase + (size_t)lo) * F;
        const int k0 = hi ? 2 : 0;
        a0.x = xp[k0];
        a0.y = xp[k0 + 1];
        const float t0 = xp[4];
        const float t1 = xp[5];
        a1.x = hi ? 0.f : t0;        // K=6,7 zero-padded
        a1.y = hi ? 0.f : t1;
    }

    // ---- GEMM: 16 N-tiles, 2 K-steps each ----
    v8f c[16];
#pragma unroll
    for (int n = 0; n < 16; ++n) {
        const int h  = n * 16 + lo;
        const int kb = hi ? 2 : 0;
        v2f b0v, b1v;
        b0v.x = sW[kb * H + h];              // B 4x16: VGPR v = K row v / v+2
        b0v.y = sW[(kb + 1) * H + h];
        const float t0 = sW[4 * H + h];
        const float t1 = sW[5 * H + h];
        b1v.x = hi ? 0.f : t0;
        b1v.y = hi ? 0.f : t1;
        v8f acc = {0.f, 0.f, 0.f, 0.f, 0.f, 0.f, 0.f, 0.f};
        acc = __builtin_amdgcn_wmma_f32_16x16x4_f32(
                  false, a0, false, b0v, (short)0, acc, false, false);
        acc = __builtin_amdgcn_wmma_f32_16x16x4_f32(
                  false, a1, false, b1v, (short)0, acc, false, false);
        c[n] = acc;
    }

    // ---- add bsum + emb (s = row & 511; no wrap inside a 16-row tile) ----
    const int sIdxBase = (int)(rowBase & (S_LEN - 1)) + 8 * hi;
#pragma unroll
    for (int n = 0; n < 16; ++n) {
        const int h = n * 16 + lo;
        const float add_h = sBsum[h];
#pragma unroll
        for (int v = 0; v < 8; ++v) {
            c[n][v] += add_h + emb[(size_t)(sIdxBase + v) * H + h];
        }
    }

    // ---- LayerNorm stats: per-VGPR tile sums, then 16-lane xor butterfly ----
    float mean[8], rstd[8];
#pragma unroll
    for (int v = 0; v < 8; ++v) {
        float s1 = 0.f, s2 = 0.f;
#pragma unroll
        for (int n = 0; n < 16; ++n) {
            const float t = c[n][v];
            s1 += t;
            s2 += t * t;
        }
#pragma unroll
        for (int m = 1; m < 16; m <<= 1) {   // masks 1,2,4,8: stay in half-wave
            s1 += __shfl_xor(s1, m, 32);
            s2 += __shfl_xor(s2, m, 32);
        }
        const float mu  = s1 * (1.0f / H);
        const float var = s2 * (1.0f / H) - mu * mu;
        mean[v] = mu;
        rstd[v] = rsqrtf(var + LN_EPS);
    }

    // ---- normalize + affine + streaming store (64B coalesced segments) ----
#pragma unroll
    for (int n = 0; n < 16; ++n) {
        const int h = n * 16 + lo;
        const float g  = sGamma[h];
        const float be = sBeta[h];
#pragma unroll
        for (int v = 0; v < 8; ++v) {
            const size_t r = rowBase + (size_t)(v + 8 * hi);
            const float val = (c[n][v] - mean[v]) * rstd[v] * g + be;
            __builtin_nontemporal_store(val, &out[r * H + h]);
        }
    }
}

extern "C" void kernel_launch(void* const* d_in, const int* in_sizes, int n_in,
                              void* d_out, int out_size, void* d_ws, size_t ws_size,
                              hipStream_t stream) {
    const float* x     = (const float*)d_in[0];
    const float* W     = (const float*)d_in[1];
    const float* b     = (const float*)d_in[2];
    const float* emb   = (const float*)d_in[3];
    const float* gamma = (const float*)d_in[4];
    const float* beta  = (const float*)d_in[5];
    float* out = (float*)d_out;

    const size_t rows = (size_t)in_sizes[0] / F;       // B*S = 1,048,576
    const int grid = (int)(rows / ROWS_PER_BLOCK);     // 8192 blocks
    stab_embed_ln_kernel<<<grid, 256, 0, stream>>>(x, W, b, emb, gamma, beta, out);
}